// SAGENet_35485019799827
// MI455X (gfx1250) — compile-verified
//
#include <hip/hip_runtime.h>

// GraphSAGE (2x SAGEConv mean + linear head) for MI455X / gfx1250.
// Aggregation via f32 global atomics (L2-resident features), dense layers via
// fp32 WMMA V_WMMA_F32_16X16X4_F32 (full precision, memory-bound anyway).

#define NNODES 50000
#define FEAT   128

typedef float v2f __attribute__((ext_vector_type(2)));
typedef float v8f __attribute__((ext_vector_type(8)));

__global__ void zero_kernel(float* __restrict__ p, long long n) {
    long long i = (long long)blockIdx.x * blockDim.x + threadIdx.x;
    if (i < n) p[i] = 0.0f;
}

__global__ void count_kernel(const int* __restrict__ dst, float* __restrict__ cnt, int E) {
    int e = blockIdx.x * blockDim.x + threadIdx.x;
    if (e < E) atomicAdd(cnt + dst[e], 1.0f);
}

// 32 lanes per edge; each lane moves a float4 chunk of the 128-wide feature row.
__global__ void scatter_kernel(const float* __restrict__ feat,
                               const int* __restrict__ src,
                               const int* __restrict__ dst,
                               float* __restrict__ agg, int E) {
    long long gid = (long long)blockIdx.x * blockDim.x + threadIdx.x;
    if (gid >= (long long)E * 32) return;
    int e   = (int)(gid >> 5);
    int col = ((int)gid & 31) << 2;
    int s = src[e];
    int d = dst[e];
    const float4 v = *reinterpret_cast<const float4*>(feat + (size_t)s * FEAT + col);
    float* ap = agg + (size_t)d * FEAT + col;
    atomicAdd(ap + 0, v.x);
    atomicAdd(ap + 1, v.y);
    atomicAdd(ap + 2, v.z);
    atomicAdd(ap + 3, v.w);
}

// out[16x16 tile] = (A1*scale) @ W1^T + A2 @ W2^T + bias, ReLU.
// One wave per tile. A is 16x4 (2 VGPRs), B is 4x16 (2 VGPRs):
//   lane<16  holds K = k0+0, k0+1 ; lane>=16 holds K = k0+2, k0+3
// A indexed by row = lane&15, B by col = lane&15 (B[k][j] = W[j][k]).
// D layout: lanes 0-15 -> rows r (VGPR r), lanes 16-31 -> rows r+8.
__global__ __launch_bounds__(128) void sage_gemm_kernel(
    const float* __restrict__ A1, const float* __restrict__ cnt,
    const float* __restrict__ A2,
    const float* __restrict__ W1, const float* __restrict__ W2,
    const float* __restrict__ bias,
    float* __restrict__ out, int nRowTiles, int outDim)
{
    const int lane     = threadIdx.x & 31;
    const int wave     = threadIdx.x >> 5;
    const int colTiles = outDim >> 4;
    const int tile     = blockIdx.x * (blockDim.x >> 5) + wave;
    if (tile >= nRowTiles * colTiles) return;   // wave-uniform: EXEC stays all-ones
    const int rowTile = tile / colTiles;
    const int colTile = tile - rowTile * colTiles;
    const int half = lane >> 4;
    const int l16  = lane & 15;
    const int row  = rowTile * 16 + l16;
    const int col  = colTile * 16 + l16;
    const int kOff = half * 2;

    float scale1 = 1.0f;
    if (cnt) scale1 = 1.0f / fmaxf(cnt[row], 1.0f);   // mean = sum / max(cnt,1)

    const float* a1row = A1 + (size_t)row * FEAT;
    const float* w1col = W1 + (size_t)col * FEAT;

    v8f acc = {};
    #pragma unroll 8
    for (int k0 = 0; k0 < FEAT; k0 += 4) {
        const int k = k0 + kOff;
        v2f a, b;
        a[0] = a1row[k]     * scale1;
        a[1] = a1row[k + 1] * scale1;
        b[0] = w1col[k];
        b[1] = w1col[k + 1];
        acc = __builtin_amdgcn_wmma_f32_16x16x4_f32(
                  false, a, false, b, (short)0, acc, false, false);
    }
    if (A2) {   // second (root/self) matmul fused into same accumulator
        const float* a2row = A2 + (size_t)row * FEAT;
        const float* w2col = W2 + (size_t)col * FEAT;
        #pragma unroll 8
        for (int k0 = 0; k0 < FEAT; k0 += 4) {
            const int k = k0 + kOff;
            v2f a, b;
            a[0] = a2row[k];
            a[1] = a2row[k + 1];
            b[0] = w2col[k];
            b[1] = w2col[k + 1];
            acc = __builtin_amdgcn_wmma_f32_16x16x4_f32(
                      false, a, false, b, (short)0, acc, false, false);
        }
    }

    const float bj = bias[col];
    #pragma unroll
    for (int r = 0; r < 8; ++r) {
        float v = acc[r] + bj;
        v = fmaxf(v, 0.0f);
        out[(size_t)(rowTile * 16 + half * 8 + r) * outDim + col] = v;
    }
}

extern "C" void kernel_launch(void* const* d_in, const int* in_sizes, int n_in,
                              void* d_out, int out_size, void* d_ws, size_t ws_size,
                              hipStream_t stream) {
    (void)n_in; (void)out_size; (void)ws_size;
    const float* x   = (const float*)d_in[0];
    const int*   ei  = (const int*)d_in[1];   // [2, E] row-major: row0=src, row1=dst
    const float* W1l = (const float*)d_in[2];
    const float* b1  = (const float*)d_in[3];
    const float* W1r = (const float*)d_in[4];
    const float* W2l = (const float*)d_in[5];
    const float* b2  = (const float*)d_in[6];
    const float* W2r = (const float*)d_in[7];
    const float* W3  = (const float*)d_in[8];
    const float* b3  = (const float*)d_in[9];

    const int N = NNODES;
    const int E = in_sizes[1] / 2;
    const int* src = ei;
    const int* dst = ei + E;

    float* agg = (float*)d_ws;                    // N*128
    float* cnt = agg + (size_t)N * FEAT;          // N
    float* h1  = cnt + N;                         // N*128
    float* h2  = h1 + (size_t)N * FEAT;           // N*128

    const long long nzAll = (long long)N * FEAT + N;   // agg+cnt contiguous
    const long long nzAgg = (long long)N * FEAT;
    const long long sw    = (long long)E * 32;
    const int rowTiles = N / 16;                       // 3125 (exact)
    const int tiles128 = rowTiles * (FEAT / 16);
    const int tiles64  = rowTiles * (64 / 16);

    // Layer 1
    zero_kernel<<<(int)((nzAll + 255) / 256), 256, 0, stream>>>(agg, nzAll);
    count_kernel<<<(E + 255) / 256, 256, 0, stream>>>(dst, cnt, E);
    scatter_kernel<<<(int)((sw + 255) / 256), 256, 0, stream>>>(x, src, dst, agg, E);
    sage_gemm_kernel<<<(tiles128 + 3) / 4, 128, 0, stream>>>(
        agg, cnt, x, W1l, W1r, b1, h1, rowTiles, FEAT);

    // Layer 2 (counts are graph-only, reuse cnt)
    zero_kernel<<<(int)((nzAgg + 255) / 256), 256, 0, stream>>>(agg, nzAgg);
    scatter_kernel<<<(int)((sw + 255) / 256), 256, 0, stream>>>(h1, src, dst, agg, E);
    sage_gemm_kernel<<<(tiles128 + 3) / 4, 128, 0, stream>>>(
        agg, cnt, h1, W2l, W2r, b2, h2, rowTiles, FEAT);

    // Head: out = relu(h2 @ W3^T + b3)
    sage_gemm_kernel<<<(tiles64 + 3) / 4, 128, 0, stream>>>(
        h2, nullptr, nullptr, W3, nullptr, b3, (float*)d_out, rowTiles, 64);
}